// HybridConvLayer_8718783611088
// MI455X (gfx1250) — compile-verified
//
#include <hip/hip_runtime.h>
#include <hip/hip_bf16.h>
#include <stdint.h>

typedef __attribute__((ext_vector_type(2))) float v2f;
typedef __attribute__((ext_vector_type(8))) float v8f;

#define NN 100000
#define NE 1600000

// ---- workspace layout (float offsets) ----
static const size_t F_DEG  = 0;                    // 100000 floats
static const size_t F_DINV = 131072;               // 100000 floats
static const size_t F_EW   = 262144;               // 1600000 floats
static const size_t F_WEFF = 2359296;              // 256*64 floats
static const size_t F_H1   = 4194304;              // 6.4M floats each
static const size_t F_H2   = F_H1 + 6400000;
static const size_t F_HT   = F_H2 + 6400000;
static const size_t F_H4   = F_HT + 6400000;

// deg[i] = 1 (self loop)
__global__ __launch_bounds__(256) void k_init_deg(float* __restrict__ deg) {
    int i = blockIdx.x * 256 + threadIdx.x;
    if (i < NN) deg[i] = 1.0f;
}

// deg[col[e]] += 1
__global__ __launch_bounds__(256) void k_accum_deg(const long long* __restrict__ col,
                                                   float* __restrict__ deg) {
    int e = blockIdx.x * 256 + threadIdx.x;
    if (e < NE) atomicAdd(&deg[(int)col[e]], 1.0f);
}

// dinv = rsqrt(deg)  (deg >= 1 always due to self loop)
__global__ __launch_bounds__(256) void k_dinv(const float* __restrict__ deg,
                                              float* __restrict__ dinv) {
    int i = blockIdx.x * 256 + threadIdx.x;
    if (i < NN) dinv[i] = rsqrtf(deg[i]);
}

// ew[e] = dinv[row[e]] * dinv[col[e]]
__global__ __launch_bounds__(256) void k_ew(const long long* __restrict__ row,
                                            const long long* __restrict__ col,
                                            const float* __restrict__ dinv,
                                            float* __restrict__ ew) {
    int e = blockIdx.x * 256 + threadIdx.x;
    if (e < NE) ew[e] = dinv[(int)row[e]] * dinv[(int)col[e]];
}

// out[i][:] = (dinv[i]^2) * h[i][:]   (self-loop term; also initializes out)
__global__ __launch_bounds__(256) void k_spmm_self(const float* __restrict__ h,
                                                   const float* __restrict__ dinv,
                                                   float* __restrict__ out) {
    int t = blockIdx.x * 256 + threadIdx.x;   // NN*16 threads, float4 each
    if (t < NN * 16) {
        int i = t >> 4;
        float d = dinv[i];
        float w = d * d;
        float4 v = ((const float4*)h)[t];
        float4 o = make_float4(v.x * w, v.y * w, v.z * w, v.w * w);
        ((float4*)out)[t] = o;
    }
}

// out[col[e]][:] += ew[e] * h[row[e]][:]   (16 threads/edge, float4 each)
__global__ __launch_bounds__(256) void k_spmm_scatter(const long long* __restrict__ row,
                                                      const long long* __restrict__ col,
                                                      const float* __restrict__ ew,
                                                      const float* __restrict__ h,
                                                      float* __restrict__ out) {
    long long t = (long long)blockIdx.x * 256 + threadIdx.x;   // NE*16 threads
    if (t < (long long)NE * 16) {
        int e = (int)(t >> 4);
        int q = (int)(t & 15);
        int r = (int)row[e];
        int c = (int)col[e];
        float w = ew[e];
        float4 v = ((const float4*)h)[r * 16 + q];
        float* o = out + (size_t)c * 64 + q * 4;
        atomicAdd(o + 0, w * v.x);
        atomicAdd(o + 1, w * v.y);
        atomicAdd(o + 2, w * v.z);
        atomicAdd(o + 3, w * v.w);
    }
}

// Fold the 6-channel concat GEMM into 4 effective 64x64 weights.
// Weff rows [0,64)=h0, [64,128)=h1, [128,192)=h2, [192,256)=h4.
__global__ __launch_bounds__(256) void k_weff(const float* __restrict__ W,
                                              float* __restrict__ Weff) {
    int t = blockIdx.x * 256 + threadIdx.x;     // 256*64 elements
    if (t < 256 * 64) {
        int r = t >> 6;          // 0..255
        int c = t & 63;
        int rad = r >> 6;        // 0..3
        int k = r & 63;
        float v;
        if (rad == 0)      v = W[(3*64 + k)*64 + c];
        else if (rad == 1) v = W[(0*64 + k)*64 + c] - W[(3*64 + k)*64 + c] + W[(4*64 + k)*64 + c];
        else if (rad == 2) v = W[(1*64 + k)*64 + c] - W[(4*64 + k)*64 + c] + W[(5*64 + k)*64 + c];
        else               v = W[(2*64 + k)*64 + c] - W[(5*64 + k)*64 + c];
        Weff[t] = v;
    }
}

// One radius's K=64 contribution: 16 k-steps x 4 n-tiles of v_wmma_f32_16x16x4_f32
__device__ __forceinline__ void wmma_radius(const float* __restrict__ hrow,   // row base (already + mrow*64)
                                            const float* __restrict__ Bs,     // LDS Weff, 256x64
                                            int rad, int koff, int bcol,
                                            v8f acc[4]) {
#pragma unroll
    for (int kk = 0; kk < 64; kk += 4) {
        v2f a;
        a.x = hrow[kk + koff];          // lanes 0-15: K=kk,kk+1 ; lanes 16-31: K=kk+2,kk+3
        a.y = hrow[kk + koff + 1];
        const float* bp = Bs + (rad * 64 + kk + koff) * 64;
#pragma unroll
        for (int j = 0; j < 4; ++j) {
            v2f b;
            b.x = bp[j * 16 + bcol];            // K row kk(+koff),   N = j*16+bcol
            b.y = bp[64 + j * 16 + bcol];       // K row kk(+koff)+1
            acc[j] = __builtin_amdgcn_wmma_f32_16x16x4_f32(
                false, a, false, b, (short)0, acc[j], false, false);
        }
    }
}

// out = relu(h0@We0 + h1@We1 + h2@We2 + h4@We3 + bias)
// 8 waves/block, each wave: 16 rows x 64 cols via 4 WMMA accumulators.
__global__ __launch_bounds__(256) void k_gemm_wmma(const float* __restrict__ h0,
                                                   const float* __restrict__ h1,
                                                   const float* __restrict__ h2,
                                                   const float* __restrict__ h4,
                                                   const float* __restrict__ Weff,
                                                   const float* __restrict__ bias,
                                                   float* __restrict__ out) {
    __shared__ float Bs[256 * 64];   // 64 KB effective-weight stage
    {
        const float4* src = (const float4*)Weff;
        float4* dst = (float4*)Bs;
        for (int i = threadIdx.x; i < 256 * 16; i += 256) dst[i] = src[i];
    }
    __syncthreads();

    int wave = threadIdx.x >> 5;
    int lane = threadIdx.x & 31;
    int m0 = (blockIdx.x * 8 + wave) * 16;
    if (m0 >= NN) return;                       // wave-uniform: EXEC stays all-ones

    int mrow = m0 + (lane & 15);
    int koff = (lane >> 4) << 1;                // 0 for lanes 0-15, 2 for lanes 16-31
    int bcol = lane & 15;

    v8f acc[4] = {};

    wmma_radius(h0 + (size_t)mrow * 64, Bs, 0, koff, bcol, acc);
    wmma_radius(h1 + (size_t)mrow * 64, Bs, 1, koff, bcol, acc);
    wmma_radius(h2 + (size_t)mrow * 64, Bs, 2, koff, bcol, acc);
    wmma_radius(h4 + (size_t)mrow * 64, Bs, 3, koff, bcol, acc);

    // C/D layout: VGPR r -> row m0 + r (+8 for lanes 16-31), col = j*16 + (lane&15)
    int rbase = m0 + ((lane >> 4) << 3);
#pragma unroll
    for (int j = 0; j < 4; ++j) {
        int c = j * 16 + bcol;
        float bv = bias[c];
#pragma unroll
        for (int r = 0; r < 8; ++r) {
            float v = acc[j][r] + bv;
            out[(size_t)(rbase + r) * 64 + c] = v > 0.0f ? v : 0.0f;
        }
    }
}

extern "C" void kernel_launch(void* const* d_in, const int* in_sizes, int n_in,
                              void* d_out, int out_size, void* d_ws, size_t ws_size,
                              hipStream_t stream) {
    const float*     x   = (const float*)d_in[0];
    const long long* ei  = (const long long*)d_in[1];   // int64 per reference
    const float*     W   = (const float*)d_in[2];
    const float*     b   = (const float*)d_in[3];
    const long long* row = ei;
    const long long* col = ei + NE;

    float* ws   = (float*)d_ws;
    float* deg  = ws + F_DEG;
    float* dinv = ws + F_DINV;
    float* ew   = ws + F_EW;
    float* Weff = ws + F_WEFF;
    float* h1   = ws + F_H1;
    float* h2   = ws + F_H2;
    float* ht   = ws + F_HT;
    float* h4   = ws + F_H4;
    float* outp = (float*)d_out;

    const int TB = 256;
    int gN  = (NN + TB - 1) / TB;
    int gE  = (NE + TB - 1) / TB;
    int gNF = (NN * 16) / TB;          // 6250
    int gEF = (NE * 16) / TB;          // 100000

    // GCN normalization
    k_init_deg <<<gN, TB, 0, stream>>>(deg);
    k_accum_deg<<<gE, TB, 0, stream>>>(col, deg);
    k_dinv     <<<gN, TB, 0, stream>>>(deg, dinv);
    k_ew       <<<gE, TB, 0, stream>>>(row, col, dinv, ew);
    k_weff     <<<(256 * 64) / TB, TB, 0, stream>>>(W, Weff);

    // h1 = P x ; h2 = P h1 ; ht = P h2 ; h4 = P ht
    k_spmm_self   <<<gNF, TB, 0, stream>>>(x,  dinv, h1);
    k_spmm_scatter<<<gEF, TB, 0, stream>>>(row, col, ew, x,  h1);
    k_spmm_self   <<<gNF, TB, 0, stream>>>(h1, dinv, h2);
    k_spmm_scatter<<<gEF, TB, 0, stream>>>(row, col, ew, h1, h2);
    k_spmm_self   <<<gNF, TB, 0, stream>>>(h2, dinv, ht);
    k_spmm_scatter<<<gEF, TB, 0, stream>>>(row, col, ew, h2, ht);
    k_spmm_self   <<<gNF, TB, 0, stream>>>(ht, dinv, h4);
    k_spmm_scatter<<<gEF, TB, 0, stream>>>(row, col, ew, ht, h4);

    // Fused concat-GEMM + bias + relu via f32 WMMA
    int gG = (NN + 127) / 128;          // 782 blocks of 8 waves x 16 rows
    k_gemm_wmma<<<gG, TB, 0, stream>>>(x, h1, h2, h4, Weff, b, outp);
}